// Interpenetration_71949292142878
// MI455X (gfx1250) — compile-verified
//
#include <hip/hip_runtime.h>
#include <hip/hip_bf16.h>
#include <stdint.h>

#define SIGMA_C 0.0001f
#define EPS_C   1e-12f

typedef __attribute__((ext_vector_type(4))) unsigned int u32x4;
typedef __attribute__((ext_vector_type(8))) int          i32x8;
typedef __attribute__((ext_vector_type(4))) int          i32x4;

// ---------------------------------------------------------------------------
// Kernel A: zero the 8-float output (harness poisons d_out).
// ---------------------------------------------------------------------------
__global__ void zero_out_kernel(float* out, int n) {
    int i = blockIdx.x * blockDim.x + threadIdx.x;
    if (i < n) out[i] = 0.0f;
}

// ---------------------------------------------------------------------------
// Kernel B: per (batch, face) precompute record:
//   q0 = {v0.x, v0.y, v0.z, n.x}
//   q1 = {v1.x, v1.y, v1.z, n.y}
//   q2 = {v2.x, v2.y, v2.z, n.z}
//   q3 = { c.x,  c.y,  c.z, r  }
// vertices[b] (V*3 floats = 125.7 KB) is staged into LDS via the Tensor Data
// Mover (one DMA per block), then faces gather from LDS.
// ---------------------------------------------------------------------------
__global__ void build_records_kernel(const float* __restrict__ vertices,
                                     const int*   __restrict__ faces,
                                     float4*      __restrict__ rec,
                                     int V, int F) {
    extern __shared__ float sv[];           // V*3 floats, dynamic LDS
    const int b = blockIdx.y;
    const float* vb = vertices + (size_t)b * (size_t)V * 3u;
    const unsigned nWords = (unsigned)V * 3u;

#if __has_builtin(__builtin_amdgcn_tensor_load_to_lds)
    if ((threadIdx.x >> 5) == 0) {          // wave 0 only issues the DMA
        const uint64_t gaddr = (uint64_t)(uintptr_t)vb;
        const unsigned laddr = (unsigned)(uintptr_t)sv;  // low 32b = LDS byte offset

        u32x4 g0;
        g0.x = 1u;                                        // count=1, user descriptor
        g0.y = laddr;                                     // lds_addr
        g0.z = (unsigned)(gaddr & 0xFFFFFFFFu);           // global_addr[31:0]
        g0.w = (unsigned)((gaddr >> 32) & 0x1FFFFFFu)     // global_addr[56:32]
             | (2u << 30);                                // type = 2 ("image")

        i32x8 g1;
        g1[0] = (int)(2u << 16);                          // data_size = 4 bytes
        g1[1] = (int)((nWords & 0xFFFFu) << 16);          // tensor_dim0[15:0]
        g1[2] = (int)((nWords >> 16) & 0xFFFFu)           // tensor_dim0[31:16]
              | (1 << 16);                                // tensor_dim1 = 1
        g1[3] = (int)((nWords & 0xFFFFu) << 16);          // tile_dim0 = V*3 dwords
        g1[4] = 1;                                        // tile_dim1 = 1
        g1[5] = (int)nWords;                              // tensor_dim0_stride[31:0]
        g1[6] = 0;                                        // stride hi / dim1_stride lo
        g1[7] = 0;

        i32x4 gz4 = {0, 0, 0, 0};                         // groups 2/3: <=2D tensor
        i32x8 gz8 = {0, 0, 0, 0, 0, 0, 0, 0};             // extra group (zero-fill)
        __builtin_amdgcn_tensor_load_to_lds(g0, g1, gz4, gz4, gz8, 0);
    }
#if __has_builtin(__builtin_amdgcn_s_wait_tensorcnt)
    __builtin_amdgcn_s_wait_tensorcnt(0);
#else
    asm volatile("s_wait_tensorcnt 0x0" ::: "memory");
#endif
#else
    for (unsigned i = threadIdx.x; i < nWords; i += blockDim.x) sv[i] = vb[i];
#endif
    __syncthreads();

    const int perBlock = (F + (int)gridDim.x - 1) / (int)gridDim.x;
    const int f0 = blockIdx.x * perBlock;
    const int f1 = min(F, f0 + perBlock);

    for (int f = f0 + (int)threadIdx.x; f < f1; f += (int)blockDim.x) {
        const int ia = faces[3 * f + 0];
        const int ib = faces[3 * f + 1];
        const int ic = faces[3 * f + 2];

        const float v0x = sv[3 * ia], v0y = sv[3 * ia + 1], v0z = sv[3 * ia + 2];
        const float v1x = sv[3 * ib], v1y = sv[3 * ib + 1], v1z = sv[3 * ib + 2];
        const float v2x = sv[3 * ic], v2y = sv[3 * ic + 1], v2z = sv[3 * ic + 2];

        const float third = 1.0f / 3.0f;
        const float cx = (v0x + v1x + v2x) * third;
        const float cy = (v0y + v1y + v2y) * third;
        const float cz = (v0z + v1z + v2z) * third;

        const float e1x = v1x - v0x, e1y = v1y - v0y, e1z = v1z - v0z;
        const float e2x = v2x - v0x, e2y = v2y - v0y, e2z = v2z - v0z;
        float nx = e1y * e2z - e1z * e2y;
        float ny = e1z * e2x - e1x * e2z;
        float nz = e1x * e2y - e1y * e2x;
        const float nl  = sqrtf(nx * nx + ny * ny + nz * nz);
        const float ninv = 1.0f / fmaxf(nl, EPS_C);
        nx *= ninv; ny *= ninv; nz *= ninv;

        const float d0 = (v0x-cx)*(v0x-cx) + (v0y-cy)*(v0y-cy) + (v0z-cz)*(v0z-cz);
        const float d1 = (v1x-cx)*(v1x-cx) + (v1y-cy)*(v1y-cy) + (v1z-cz)*(v1z-cz);
        const float d2 = (v2x-cx)*(v2x-cx) + (v2y-cy)*(v2y-cy) + (v2z-cz)*(v2z-cz);
        const float r  = sqrtf(fmaxf(fmaxf(d0, d1), d2));

        float4* o = rec + ((size_t)b * (size_t)F + (size_t)f) * 4u;
        o[0] = make_float4(v0x, v0y, v0z, nx);
        o[1] = make_float4(v1x, v1y, v1z, ny);
        o[2] = make_float4(v2x, v2y, v2z, nz);
        o[3] = make_float4(cx,  cy,  cz,  r );
    }
}

// ---------------------------------------------------------------------------
// Kernel C: per pair, gather two 64B records, evaluate cone psi for 6 points,
// wave32-shuffle + LDS reduce, one atomicAdd per block per batch.
// ---------------------------------------------------------------------------
__device__ __forceinline__ float cone_psi2(float px, float py, float pz,
                                           float cx, float cy, float cz,
                                           float nx, float ny, float nz,
                                           float inv_r) {
    const float ux = px - cx, uy = py - cy, uz = pz - cz;
    const float h  = ux * nx + uy * ny + uz * nz;
    const float wx = ux - h * nx, wy = uy - h * ny, wz = uz - h * nz;
    const float rho = sqrtf(wx * wx + wy * wy + wz * wz);
    const float radial = fmaxf(1.0f - rho * inv_r, 0.0f);
    const float axial  = fmaxf(SIGMA_C - h, 0.0f);
    const float psi = radial * axial;
    return psi * psi;
}

__global__ void pair_loss_kernel(const int2*   __restrict__ coll,
                                 const float4* __restrict__ rec,
                                 float*        __restrict__ out,
                                 int C, int F) {
    const int b = blockIdx.y;
    const int2* cb = coll + (size_t)b * (size_t)C;

    float acc = 0.0f;
    for (int c = blockIdx.x * blockDim.x + threadIdx.x; c < C;
         c += gridDim.x * blockDim.x) {
        const int2 ij = cb[c];
        if ((ij.x | ij.y) >= 0) {               // both indices non-negative
            const float4* R = rec + ((size_t)b * (size_t)F + (size_t)ij.x) * 4u;
            const float4* I = rec + ((size_t)b * (size_t)F + (size_t)ij.y) * 4u;
            const float4 r0 = R[0], r1 = R[1], r2 = R[2], r3 = R[3];
            const float4 q0 = I[0], q1 = I[1], q2 = I[2], q3 = I[3];
            const float invR = 1.0f / fmaxf(r3.w, EPS_C);
            const float invI = 1.0f / fmaxf(q3.w, EPS_C);
            // psi_i: intruder vertices vs receiver frame
            acc += cone_psi2(q0.x, q0.y, q0.z, r3.x, r3.y, r3.z, r0.w, r1.w, r2.w, invR);
            acc += cone_psi2(q1.x, q1.y, q1.z, r3.x, r3.y, r3.z, r0.w, r1.w, r2.w, invR);
            acc += cone_psi2(q2.x, q2.y, q2.z, r3.x, r3.y, r3.z, r0.w, r1.w, r2.w, invR);
            // psi_r: receiver vertices vs intruder frame
            acc += cone_psi2(r0.x, r0.y, r0.z, q3.x, q3.y, q3.z, q0.w, q1.w, q2.w, invI);
            acc += cone_psi2(r1.x, r1.y, r1.z, q3.x, q3.y, q3.z, q0.w, q1.w, q2.w, invI);
            acc += cone_psi2(r2.x, r2.y, r2.z, q3.x, q3.y, q3.z, q0.w, q1.w, q2.w, invI);
        }
    }

    // wave32 reduction
    for (int off = 16; off > 0; off >>= 1)
        acc += __shfl_xor(acc, off, 32);

    __shared__ float wsum[8];                   // 256 threads = 8 waves
    const int lane = threadIdx.x & 31;
    const int wid  = threadIdx.x >> 5;
    if (lane == 0) wsum[wid] = acc;
    __syncthreads();
    if (threadIdx.x == 0) {
        float t = 0.0f;
        const int nw = ((int)blockDim.x + 31) >> 5;
        for (int i = 0; i < nw; ++i) t += wsum[i];
        atomicAdd(out + b, t);
    }
}

// ---------------------------------------------------------------------------
extern "C" void kernel_launch(void* const* d_in, const int* in_sizes, int n_in,
                              void* d_out, int out_size, void* d_ws, size_t ws_size,
                              hipStream_t stream) {
    const float* vertices = (const float*)d_in[0];   // [B, V, 3] f32
    const int*   faces    = (const int*)  d_in[1];   // [F, 3] i32
    const int*   coll     = (const int*)  d_in[2];   // [B, C, 2] i32
    float*       out      = (float*)d_out;           // [B] f32

    const int B = out_size;                          // 8
    const int V = in_sizes[0] / (B * 3);             // 10475
    const int F = in_sizes[1] / 3;                   // 20908
    const int C = in_sizes[2] / (B * 2);             // 262144

    float4* rec = (float4*)d_ws;                     // B*F*4 float4 = ~10.7 MB

    zero_out_kernel<<<1, 32, 0, stream>>>(out, B);

    const size_t ldsBytes = (size_t)V * 3u * sizeof(float);   // 125.7 KB < 320 KB
    (void)hipFuncSetAttribute(reinterpret_cast<const void*>(build_records_kernel),
                              hipFuncAttributeMaxDynamicSharedMemorySize,
                              (int)ldsBytes);
    dim3 gridB(32, B);
    build_records_kernel<<<gridB, 256, ldsBytes, stream>>>(vertices, faces, rec, V, F);

    dim3 gridC(256, B);
    pair_loss_kernel<<<gridC, 256, 0, stream>>>((const int2*)coll, rec, out, C, F);
}